// GraphDecoder_67259187855553
// MI455X (gfx1250) — compile-verified
//
#include <hip/hip_runtime.h>
#include <hip/hip_bf16.h>

#define LATENT    256
#define NODE_DIM  128
#define HIDDEN    64
#define NNODES    2048

typedef __attribute__((ext_vector_type(2))) float v2f;
typedef __attribute__((ext_vector_type(8))) float v8f;

// ---------------------------------------------------------------------------
// nodes[2048,128] = broadcast(z)[2048,256] @ W_node[256,128] + b_node
// One wave32 per 16x16 C tile; V_WMMA_F32_16X16X4_F32, K-loop of 64 steps.
// A-matrix rows are all equal to z (broadcast), so the A fragment is loaded
// straight from z with the ISA 16x4 f32 A layout:
//   lanes 0-15: M=lane, VGPR0=K0, VGPR1=K0+1 ; lanes 16-31: K0+=2
// B fragment (4x16): VGPR0 -> k = 2*half, VGPR1 -> k = 2*half+1, N = lane&15.
// C/D: VGPR v, lane l -> M = v + 8*(l>>4), N = l&15.
// ---------------------------------------------------------------------------
__global__ __launch_bounds__(32)
void node_gemm_kernel(const float* __restrict__ z,
                      const float* __restrict__ Wn,
                      const float* __restrict__ bn,
                      float* __restrict__ nodes) {
    const int lane = threadIdx.x;
    const int half = lane >> 4;     // 0 or 1
    const int l16  = lane & 15;
    const int rowBase = blockIdx.x * 16;   // 0..2047 step 16
    const int colBase = blockIdx.y * 16;   // 0..127  step 16

    v8f acc = {};
    #pragma unroll 4
    for (int kk = 0; kk < LATENT; kk += 4) {
        const int k0 = kk + 2 * half;
        v2f afrag, bfrag;
        afrag.x = z[k0 + 0];                                // A[M][k] = z[k]
        afrag.y = z[k0 + 1];
        bfrag.x = Wn[(k0 + 0) * NODE_DIM + colBase + l16];
        bfrag.y = Wn[(k0 + 1) * NODE_DIM + colBase + l16];
        acc = __builtin_amdgcn_wmma_f32_16x16x4_f32(false, afrag, false, bfrag,
                                                    (short)0, acc, false, false);
    }

    #pragma unroll
    for (int v = 0; v < 8; ++v) {
        const int M = v + 8 * half;
        const int N = l16;
        nodes[(rowBase + M) * NODE_DIM + colBase + N] = acc[v] + bn[colBase + N];
    }
}

// ---------------------------------------------------------------------------
// ap[2048,64] = nodes @ W1a + b1   (b1 folded in here)
// bb[2048,64] = nodes @ W1b
// One wave32 per 16x16 tile of BOTH outputs; shared A fragment, two WMMA accs.
// ---------------------------------------------------------------------------
__global__ __launch_bounds__(32)
void ab_gemm_kernel(const float* __restrict__ nodes,
                    const float* __restrict__ W1a,
                    const float* __restrict__ W1b,
                    const float* __restrict__ b1,
                    float* __restrict__ ap,
                    float* __restrict__ bb) {
    const int lane = threadIdx.x;
    const int half = lane >> 4;
    const int l16  = lane & 15;
    const int rowBase = blockIdx.x * 16;   // 0..2047 step 16
    const int colBase = blockIdx.y * 16;   // 0..63   step 16

    const float* arow = nodes + (rowBase + l16) * NODE_DIM;

    v8f acca = {};
    v8f accb = {};
    #pragma unroll 4
    for (int kk = 0; kk < NODE_DIM; kk += 4) {
        const int k0 = kk + 2 * half;
        v2f afrag, bfa, bfb;
        afrag.x = arow[k0 + 0];
        afrag.y = arow[k0 + 1];
        bfa.x = W1a[(k0 + 0) * HIDDEN + colBase + l16];
        bfa.y = W1a[(k0 + 1) * HIDDEN + colBase + l16];
        bfb.x = W1b[(k0 + 0) * HIDDEN + colBase + l16];
        bfb.y = W1b[(k0 + 1) * HIDDEN + colBase + l16];
        acca = __builtin_amdgcn_wmma_f32_16x16x4_f32(false, afrag, false, bfa,
                                                     (short)0, acca, false, false);
        accb = __builtin_amdgcn_wmma_f32_16x16x4_f32(false, afrag, false, bfb,
                                                     (short)0, accb, false, false);
    }

    #pragma unroll
    for (int v = 0; v < 8; ++v) {
        const int M = v + 8 * half;
        const int N = l16;
        const int o = (rowBase + M) * HIDDEN + colBase + N;
        ap[o] = acca[v] + b1[colBase + N];
        bb[o] = accb[v];
    }
}

// ---------------------------------------------------------------------------
// Edge MLP + symmetrize + sigmoid. ReLU between the (a_i + b_j + b1) broadcast
// and the W2 contraction prevents a matmul formulation -> VALU kernel.
// One 256-thread block per UPPER-triangular 16x16 tile pair (I<=J): computes
// d_ij and d_ji for every pair, so each ordered pair's dot is done exactly
// once; writes tile (I,J) and its mirror (J,I).
// adj[i,j] = sigmoid(0.5*(d_ij + d_ji) + b2)
// ---------------------------------------------------------------------------
__global__ __launch_bounds__(256)
void edge_kernel(const float* __restrict__ ap,   // a + b1
                 const float* __restrict__ bb,   // b
                 const float* __restrict__ W2,
                 const float* __restrict__ b2,
                 float* __restrict__ adj) {
    __shared__ float sApI[16][HIDDEN + 4];
    __shared__ float sBbJ[16][HIDDEN + 4];
    __shared__ float sApJ[16][HIDDEN + 4];
    __shared__ float sBbI[16][HIDDEN + 4];
    __shared__ float sW2[HIDDEN];

    // decode linear block id -> (I, J), I <= J  (scalar, uniform)
    int p = blockIdx.x;
    int I = 0, rem = NNODES / 16;
    while (p >= rem) { p -= rem; ++I; --rem; }
    const int J  = I + p;
    const int Ib = I * 16, Jb = J * 16;

    const int t  = threadIdx.x;
    const int r  = t >> 4;
    const int c4 = (t & 15) * 4;

    // stage 4 x (16 rows x 64 f32) into LDS, one float4 per thread per array
    *(float4*)&sApI[r][c4] = *(const float4*)&ap[(Ib + r) * HIDDEN + c4];
    *(float4*)&sBbJ[r][c4] = *(const float4*)&bb[(Jb + r) * HIDDEN + c4];
    *(float4*)&sApJ[r][c4] = *(const float4*)&ap[(Jb + r) * HIDDEN + c4];
    *(float4*)&sBbI[r][c4] = *(const float4*)&bb[(Ib + r) * HIDDEN + c4];
    if (t < 16) *(float4*)&sW2[t * 4] = *(const float4*)&W2[t * 4];
    __syncthreads();

    const int ti = t >> 4;
    const int tj = t & 15;

    float d1 = 0.f;   // dot for ordered pair (i = Ib+ti, j = Jb+tj)
    float d2 = 0.f;   // dot for ordered pair (j, i)
    #pragma unroll
    for (int h = 0; h < HIDDEN; h += 4) {
        const float4 aI = *(const float4*)&sApI[ti][h];
        const float4 bJ = *(const float4*)&sBbJ[tj][h];
        const float4 aJ = *(const float4*)&sApJ[tj][h];
        const float4 bI = *(const float4*)&sBbI[ti][h];
        const float4 w  = *(const float4*)&sW2[h];
        d1 = fmaf(fmaxf(aI.x + bJ.x, 0.f), w.x, d1);
        d1 = fmaf(fmaxf(aI.y + bJ.y, 0.f), w.y, d1);
        d1 = fmaf(fmaxf(aI.z + bJ.z, 0.f), w.z, d1);
        d1 = fmaf(fmaxf(aI.w + bJ.w, 0.f), w.w, d1);
        d2 = fmaf(fmaxf(aJ.x + bI.x, 0.f), w.x, d2);
        d2 = fmaf(fmaxf(aJ.y + bI.y, 0.f), w.y, d2);
        d2 = fmaf(fmaxf(aJ.z + bI.z, 0.f), w.z, d2);
        d2 = fmaf(fmaxf(aJ.w + bI.w, 0.f), w.w, d2);
    }

    const float logit = 0.5f * (d1 + d2) + b2[0];
    const float s = 1.0f / (1.0f + __expf(-logit));

    adj[(Ib + ti) * NNODES + (Jb + tj)] = s;
    adj[(Jb + tj) * NNODES + (Ib + ti)] = s;   // mirror (diag tiles: same value)
}

// ---------------------------------------------------------------------------
extern "C" void kernel_launch(void* const* d_in, const int* in_sizes, int n_in,
                              void* d_out, int out_size, void* d_ws, size_t ws_size,
                              hipStream_t stream) {
    (void)in_sizes; (void)n_in; (void)out_size; (void)ws_size;

    const float* z   = (const float*)d_in[0];
    const float* Wn  = (const float*)d_in[1];
    const float* bn  = (const float*)d_in[2];
    const float* W1a = (const float*)d_in[3];
    const float* W1b = (const float*)d_in[4];
    const float* b1  = (const float*)d_in[5];
    const float* W2  = (const float*)d_in[6];
    const float* b2  = (const float*)d_in[7];
    // d_in[8] = true_num_nodes (static 2048 per reference module constants)

    float* nodes = (float*)d_out;                       // [2048,128]
    float* adj   = nodes + NNODES * NODE_DIM;           // [2048,2048]

    float* ap = (float*)d_ws;                           // a + b1  [2048,64]
    float* bbuf = ap + NNODES * HIDDEN;                 // b       [2048,64]

    node_gemm_kernel<<<dim3(NNODES / 16, NODE_DIM / 16), 32, 0, stream>>>(
        z, Wn, bn, nodes);

    ab_gemm_kernel<<<dim3(NNODES / 16, HIDDEN / 16), 32, 0, stream>>>(
        nodes, W1a, W1b, b1, ap, bbuf);

    const int NT = NNODES / 16;                          // 128 tiles/side
    edge_kernel<<<NT * (NT + 1) / 2, 256, 0, stream>>>(  // 8256 tile pairs
        ap, bbuf, W2, b2, adj);
}